// GraphConvolution_84490596647560
// MI455X (gfx1250) — compile-verified
//
#include <hip/hip_runtime.h>
#include <stdint.h>

#define TILE 512   // edges staged per block via the Tensor Data Mover
#define HD   256   // H*D floats per node (4 heads * 64)

typedef uint32_t v4u __attribute__((ext_vector_type(4)));
typedef int      v8i __attribute__((ext_vector_type(8)));
typedef int      v4i __attribute__((ext_vector_type(4)));

// Build a Tensor-DMA descriptor (D#) for a small 2D tile:
//   tile = rows x tile_x elements of (1 << ds_code) bytes,
//   tensor extent in X = tensor_x (reads with X >= tensor_x return zero),
//   global row pitch = row_stride_elems elements,
//   packed contiguously into LDS at lds_off.
// Issues TENSOR_LOAD_TO_LDS (tracked by TENSORcnt).
__device__ __forceinline__ void tdm_load_to_lds(uint32_t lds_off,
                                                const void* gptr,
                                                uint32_t tile_x,
                                                uint32_t tensor_x,
                                                uint32_t rows,
                                                uint64_t row_stride_elems,
                                                uint32_t ds_code) {
  const uint64_t ga = (uint64_t)gptr;
  v4u g0;
  g0.x = 1u;                                   // count=1 valid user descriptor
  g0.y = lds_off;                              // lds_addr (bytes)
  g0.z = (uint32_t)ga;                         // global_addr[31:0]
  g0.w = (uint32_t)(ga >> 32) | (2u << 30);    // global_addr[56:32] | type=2
  v8i g1;
  g1[0] = (int)(ds_code << 16);                // wg_mask=0, data_size, no iterate/pad
  g1[1] = (int)((tensor_x & 0xFFFFu) << 16);   // tensor_dim0[15:0] @ bit 48
  g1[2] = (int)((tensor_x >> 16) | ((rows & 0xFFFFu) << 16)); // tdim0 hi | tensor_dim1 lo
  g1[3] = (int)((tile_x & 0xFFFFu) << 16);     // tensor_dim1 hi=0 | tile_dim0
  g1[4] = (int)(rows > 1u ? rows : 0u);        // tile_dim1 (0 = unused), tile_dim2=0
  g1[5] = (int)(uint32_t)row_stride_elems;     // tensor_dim0_stride[31:0]
  g1[6] = (int)(uint32_t)((row_stride_elems >> 32) & 0xFFFFu); // stride hi; dim1_stride=0
  g1[7] = 0;
  v4i z4 = {0, 0, 0, 0};
#if defined(__clang_major__) && (__clang_major__ >= 23)
  v8i z8 = {0, 0, 0, 0, 0, 0, 0, 0};
  __builtin_amdgcn_tensor_load_to_lds(g0, g1, z4, z4, z8, 0);
#else
  __builtin_amdgcn_tensor_load_to_lds(g0, g1, z4, z4, 0);
#endif
}

// Hot kernel first in the file so the disasm snippet shows it.
//
// TDM-staged gather/scatter. One block = one tile of TILE edges.
// Wave w (32 lanes) owns edges k = w, w+8, ... ; each lane owns 8 of the
// 256 floats (two float4 gathers + 8 non-returning f32 atomics).
__global__ __launch_bounds__(256)
void gcn_scatter(const float* __restrict__ x,
                 const float* __restrict__ ew,
                 const long long* __restrict__ eidx,  // [2*E]: rows then cols
                 const float* __restrict__ deg,
                 float* __restrict__ out,
                 int E) {
  __shared__ long long s_rc[2 * TILE];  // [0,TILE): row  [TILE,2*TILE): col
  __shared__ float     s_w[TILE];

  const int base = blockIdx.x * TILE;
  int cnt = E - base;
  if (cnt <= 0) return;
  if (cnt > TILE) cnt = TILE;

  if (threadIdx.x == 0) {
    // rows+cols: one 2-row i64 tile, global pitch = E elements; lands as
    // s_rc[0..TILE) = row, s_rc[TILE..2*TILE) = col. tensor_dim0 = cnt so a
    // partial tail tile reads zeros instead of past the buffer end.
    tdm_load_to_lds((uint32_t)(uintptr_t)(void*)s_rc,
                    eidx + base, (uint32_t)TILE, (uint32_t)cnt, 2u,
                    (uint64_t)E, 3u);
    // weights: one f32 row
    tdm_load_to_lds((uint32_t)(uintptr_t)(void*)s_w,
                    ew + base, (uint32_t)TILE, (uint32_t)cnt, 1u,
                    (uint64_t)TILE, 2u);
  }
  __builtin_amdgcn_s_wait_tensorcnt(0);  // waves with TENSORcnt==0 pass through
  __syncthreads();

  const int lane = threadIdx.x & 31;
  const int wid  = threadIdx.x >> 5;

  for (int k = wid; k < cnt; k += 8) {
    const int   r  = (int)s_rc[k];
    const int   c  = (int)s_rc[TILE + k];
    const float w  = s_w[k];
    const float dc = deg[c];
    const float dr = deg[r];
    float val = w * rsqrtf(dc) * rsqrtf(dr);
    val = (dr > 0.0f && dc > 0.0f) ? val : 0.0f;  // nan_to_num(inf/nan -> 0)

    // Prefetch the next edge's source row for this wave (global_prefetch_b8).
    if (k + 8 < cnt) {
      const int rn = (int)s_rc[k + 8];
      __builtin_prefetch(x + (size_t)rn * HD + lane * 8, 0, 3);
    }

    const float4* __restrict__ xr = (const float4*)(x + (size_t)r * HD);
    float* __restrict__ orow = out + (size_t)c * HD;

#pragma unroll
    for (int half = 0; half < 2; ++half) {
      float4 v = xr[half * 32 + lane];          // global_load_b128
      const int fo = (half * 32 + lane) * 4;
      __hip_atomic_fetch_add(&orow[fo + 0], v.x * val, __ATOMIC_RELAXED, __HIP_MEMORY_SCOPE_AGENT);
      __hip_atomic_fetch_add(&orow[fo + 1], v.y * val, __ATOMIC_RELAXED, __HIP_MEMORY_SCOPE_AGENT);
      __hip_atomic_fetch_add(&orow[fo + 2], v.z * val, __ATOMIC_RELAXED, __HIP_MEMORY_SCOPE_AGENT);
      __hip_atomic_fetch_add(&orow[fo + 3], v.w * val, __ATOMIC_RELAXED, __HIP_MEMORY_SCOPE_AGENT);
    }
  }
}

// Zero the output accumulator and the degree array.
__global__ void gcn_zero(float* __restrict__ out, int n_out,
                         float* __restrict__ deg, int n_deg) {
  const int stride = gridDim.x * blockDim.x;
  for (int i = blockIdx.x * blockDim.x + threadIdx.x; i < n_out; i += stride)
    out[i] = 0.0f;
  for (int i = blockIdx.x * blockDim.x + threadIdx.x; i < n_deg; i += stride)
    deg[i] = 0.0f;
}

// In-degree over col (float counts are exact up to 2^24).
__global__ void gcn_degree(const long long* __restrict__ col,
                           float* __restrict__ deg, int E) {
  const int i = blockIdx.x * blockDim.x + threadIdx.x;
  if (i < E) {
    __hip_atomic_fetch_add(&deg[(int)col[i]], 1.0f,
                           __ATOMIC_RELAXED, __HIP_MEMORY_SCOPE_AGENT);
  }
}

extern "C" void kernel_launch(void* const* d_in, const int* in_sizes, int n_in,
                              void* d_out, int out_size, void* d_ws, size_t ws_size,
                              hipStream_t stream) {
  const float*     x    = (const float*)d_in[0];      // [N,H,D] f32
  const float*     ew   = (const float*)d_in[1];      // [E] f32
  const long long* eidx = (const long long*)d_in[2];  // [2,E] i64
  float* out = (float*)d_out;                         // [N,H,D] f32
  float* deg = (float*)d_ws;                          // N floats of scratch

  const int E = in_sizes[1];
  const int N = in_sizes[0] / HD;

  gcn_zero<<<2048, 256, 0, stream>>>(out, out_size, deg, N);
  gcn_degree<<<(E + 255) / 256, 256, 0, stream>>>(eidx + E, deg, E);

  const int nTiles = (E + TILE - 1) / TILE;
  gcn_scatter<<<nTiles, 256, 0, stream>>>(x, ew, eidx, deg, out, E);
}